// MultiBoxLoss_87299505258499
// MI455X (gfx1250) — compile-verified
//
#include <hip/hip_runtime.h>
#include <math.h>

#define B_  64
#define P_  8732
#define C_  81
#define M_  16
#define BP_ (B_ * P_)
#define THRESH 0.75f

typedef __attribute__((ext_vector_type(2))) float v2f;
typedef __attribute__((ext_vector_type(8))) float v8f;
typedef unsigned long long u64;
typedef unsigned int u32;

#if __has_builtin(__builtin_amdgcn_global_load_async_to_lds_b128)
#define HAVE_ASYNC_LDS 1
typedef int v4i_ __attribute__((vector_size(4 * sizeof(int))));
typedef __attribute__((address_space(1))) v4i_ gv4i;   // global-addrspace int4
typedef __attribute__((address_space(3))) v4i_ lv4i;   // LDS-addrspace int4
#else
#define HAVE_ASYNC_LDS 0
#endif

__device__ __forceinline__ void wait_asynccnt0() {
#if __has_builtin(__builtin_amdgcn_s_wait_asynccnt)
  __builtin_amdgcn_s_wait_asynccnt(0);
#else
  asm volatile("s_wait_asynccnt 0" ::: "memory");
#endif
}

// Exact 32-lane f32 sum on the matrix pipe:
// A(16x4): lane m (m<16) holds A[m][0]=x; lane m+16 holds A[m][2]=x; other cols 0.
// B(4x16) = ones  =>  D[m][n] = x_m + x_{m+16} for all n.
// D VGPR r: lanes 0-15 -> row r, lanes 16-31 -> row r+8. Sum 8 VGPRs + shfl_xor(16) = total.
__device__ __forceinline__ float wave_sum_wmma(float x) {
  v2f a; a.x = x;    a.y = 0.0f;
  v2f b; b.x = 1.0f; b.y = 1.0f;
  v8f c = {};
  c = __builtin_amdgcn_wmma_f32_16x16x4_f32(false, a, false, b, (short)0, c, false, false);
  float s = c[0] + c[1] + c[2] + c[3] + c[4] + c[5] + c[6] + c[7];
  s += __shfl_xor(s, 16, 32);
  return s; // all lanes hold full-wave sum
}

__global__ void init_ws(u64* bestp, int* npos, float* acc) {
  int i = blockIdx.x * blockDim.x + threadIdx.x;
  if (i < B_ * M_) bestp[i] = 0ull;
  if (i < B_)      npos[i]  = 0;
  if (i < 8)       acc[i]   = 0.0f;
}

// Per (b,p): IoU against 16 boxes; per-prior best object; per-object best prior via packed atomicMax.
__global__ void overlap_kernel(const float* __restrict__ boxes,   // (B,M,4) xy
                               const float* __restrict__ priors,  // (P,4) cxcy
                               float* __restrict__ ovl, int* __restrict__ obj,
                               u64* __restrict__ bestp) {
  __shared__ float bs[M_ * 4];
  __shared__ u64 wred[8][M_];
  int b = blockIdx.y, tid = threadIdx.x;
  if (tid < M_ * 4) bs[tid] = boxes[b * M_ * 4 + tid];
  __syncthreads();
  int p = blockIdx.x * 256 + tid;
  bool valid = p < P_;
  int pc = valid ? p : (P_ - 1);
  float pcx = priors[pc*4+0], pcy = priors[pc*4+1], pw = priors[pc*4+2], ph = priors[pc*4+3];
  float px1 = pcx - 0.5f*pw, py1 = pcy - 0.5f*ph, px2 = pcx + 0.5f*pw, py2 = pcy + 0.5f*ph;
  float pa = pw * ph;
  float ovm[M_];
  float best = -1.0f; int bi = 0;
  for (int m = 0; m < M_; ++m) {
    float x1 = bs[m*4+0], y1 = bs[m*4+1], x2 = bs[m*4+2], y2 = bs[m*4+3];
    float iw = fmaxf(fminf(px2, x2) - fmaxf(px1, x1), 0.0f);
    float ih = fmaxf(fminf(py2, y2) - fmaxf(py1, y1), 0.0f);
    float inter = iw * ih;
    float uni = (x2 - x1) * (y2 - y1) + pa - inter;
    float ov = inter / uni;
    ovm[m] = ov;
    if (ov > best) { best = ov; bi = m; }   // strict > keeps first max (jnp.argmax)
  }
  if (valid) { ovl[b * P_ + p] = best; obj[b * P_ + p] = bi; }
  int wid = tid >> 5, lane = tid & 31;
  for (int m = 0; m < M_; ++m) {
    u64 pk = valid ? (((u64)__float_as_uint(ovm[m]) << 32) | (u32)(0xFFFFFFFFu - (u32)p)) : 0ull;
    for (int off = 16; off; off >>= 1) {
      u32 lo = (u32)pk, hi = (u32)(pk >> 32);
      lo = __shfl_xor(lo, off, 32);
      hi = __shfl_xor(hi, off, 32);
      u64 o = ((u64)hi << 32) | lo;
      pk = o > pk ? o : pk;
    }
    if (lane == 0) wred[wid][m] = pk;
  }
  __syncthreads();
  if (tid < M_) {
    u64 mx = 0ull;
    for (int w = 0; w < 8; ++w) mx = wred[w][tid] > mx ? wred[w][tid] : mx;
    atomicMax(&bestp[b * M_ + tid], mx);
  }
}

__global__ void override_kernel(float* ovl, int* obj, const u64* __restrict__ bestp) {
  int i = blockIdx.x * blockDim.x + threadIdx.x;
  if (i >= B_ * M_) return;
  int b = i / M_, m = i % M_;
  int p = (int)(0xFFFFFFFFu - (u32)(bestp[i] & 0xFFFFFFFFull));
  obj[b * P_ + p] = m;
  ovl[b * P_ + p] = 1.0f;
}

// Label each prior, SSD-encode matched box, SmoothL1 vs predicted_locs; block-reduce via WMMA.
__global__ void match_loc_kernel(const float* __restrict__ ploc,   // (B,P,4)
                                 const float* __restrict__ boxes,  // (B,M,4)
                                 const int*   __restrict__ labels, // (B,M)
                                 const float* __restrict__ priors, // (P,4)
                                 const float* __restrict__ ovl,
                                 const int*   __restrict__ obj,
                                 int* __restrict__ tc,
                                 int* __restrict__ npos,
                                 float* __restrict__ acc) {
  __shared__ float bs[M_ * 4];
  __shared__ int ls[M_];
  __shared__ float wpart[8];
  __shared__ int wcnt[8];
  int b = blockIdx.y, tid = threadIdx.x;
  if (tid < M_ * 4) bs[tid] = boxes[b * M_ * 4 + tid];
  if (tid < M_)     ls[tid] = labels[b * M_ + tid];
  __syncthreads();
  int p = blockIdx.x * 256 + tid;
  bool valid = p < P_;
  int pc = valid ? p : (P_ - 1);
  int o = obj[b * P_ + pc];
  float ov = ovl[b * P_ + pc];
  int lbl = ls[o];
  if (ov < THRESH) lbl = 0;
  if (valid) tc[b * P_ + p] = lbl;
  bool pos = valid && (lbl != 0);
  float x1 = bs[o*4+0], y1 = bs[o*4+1], x2 = bs[o*4+2], y2 = bs[o*4+3];
  float cx = 0.5f*(x1+x2), cy = 0.5f*(y1+y2), w = x2 - x1, h = y2 - y1;
  float pcx = priors[pc*4+0], pcy = priors[pc*4+1], pw = priors[pc*4+2], ph = priors[pc*4+3];
  float g[4];
  g[0] = (cx - pcx) * 10.0f / pw;
  g[1] = (cy - pcy) * 10.0f / ph;
  g[2] = logf(w / pw) * 5.0f;
  g[3] = logf(h / ph) * 5.0f;
  const float* pl = ploc + ((size_t)b * P_ + (size_t)pc) * 4;
  float s = 0.0f;
  for (int k = 0; k < 4; ++k) {
    float d = __builtin_nontemporal_load(&pl[k]) - g[k];   // one-shot stream: NT hint
    float ad = fabsf(d);
    s += (ad < 1.0f) ? 0.5f * d * d : ad - 0.5f;
  }
  float wsum = wave_sum_wmma(pos ? s : 0.0f);   // EXEC all-ones: no divergence here
  u64 bal = __ballot(pos);
  int wid = tid >> 5, lane = tid & 31;
  if (lane == 0) { wpart[wid] = wsum; wcnt[wid] = __popcll(bal); }
  __syncthreads();
  if (tid == 0) {
    float tot = 0.0f; int c = 0;
    for (int w2 = 0; w2 < 8; ++w2) { tot += wpart[w2]; c += wcnt[w2]; }
    atomicAdd(&acc[0], tot);
    atomicAdd(&npos[b], c);
  }
}

// One wave per (b,p) row: logsumexp over C=81, CE at true class. Sum-of-exp via WMMA.
// scores is a 181MB single-use stream -> non-temporal loads keep L2 for reused data.
__global__ void ce_kernel(const float* __restrict__ scores, // (B,P,C)
                          const int* __restrict__ tc,
                          float* __restrict__ ceneg,
                          float* __restrict__ acc) {
  __shared__ float wpart[8];
  int tid = threadIdx.x, wid = tid >> 5, lane = tid & 31;
  long long row = (long long)blockIdx.x * 8 + wid;   // grid = BP_/8 exactly
  const float* s = scores + (size_t)row * C_;
  float v0 = __builtin_nontemporal_load(&s[lane]);
  float v1 = __builtin_nontemporal_load(&s[lane + 32]);
  float v2 = (lane < C_ - 64) ? __builtin_nontemporal_load(&s[lane + 64]) : -3.402823466e38f;
  float mx = fmaxf(fmaxf(v0, v1), v2);
  for (int off = 16; off; off >>= 1) mx = fmaxf(mx, __shfl_xor(mx, off, 32));
  float e = expf(v0 - mx) + expf(v1 - mx) + ((lane < C_ - 64) ? expf(v2 - mx) : 0.0f);
  float sum = wave_sum_wmma(e);                      // full wave active
  float lse = logf(sum) + mx;
  int lbl = tc[row];                                 // wave-uniform
  float pick = (lbl < 32) ? v0 : ((lbl < 64) ? v1 : v2);
  float sl = __shfl(pick, lbl & 31, 32);
  float ce = lse - sl;
  bool pos = (lbl != 0);
  if (lane == 0) {
    ceneg[row] = pos ? 0.0f : ce;
    wpart[wid] = pos ? ce : 0.0f;
  }
  __syncthreads();
  if (tid == 0) {
    float tot = 0.0f;
    for (int w = 0; w < 8; ++w) tot += wpart[w];
    atomicAdd(&acc[1], tot);
  }
}

// Exact sum of top-K negatives per row: binary search the K-th largest over float bit patterns.
// The 35KB row is staged into LDS via the CDNA5 async global->LDS path when available.
__global__ void hardneg_kernel(const float* __restrict__ ceneg,
                               const int* __restrict__ npos,
                               float* __restrict__ acc) {
  __shared__ alignas(16) float sv[P_];      // 34928 B (P_ = 4*2183, exact B128 chunks)
  __shared__ int scnt;
  __shared__ float ssum;
  __shared__ float wpart[8];
  int b = blockIdx.x, tid = threadIdx.x;
  const float* src = ceneg + (size_t)b * P_;
#if HAVE_ASYNC_LDS
  for (int ch = tid; ch < P_ / 4; ch += 256) {
    __builtin_amdgcn_global_load_async_to_lds_b128(
        (gv4i*)(const_cast<float*>(src) + ch * 4),
        (lv4i*)(&sv[ch * 4]), 0, 0);
  }
  wait_asynccnt0();
#else
  for (int i = tid; i < P_; i += 256) sv[i] = src[i];
#endif
  long long K = 3LL * (long long)npos[b];
  __syncthreads();
  if (K <= 0) return;                       // uniform across block
  if (K >= P_) {
    float ls = 0.0f;
    for (int i = tid; i < P_; i += 256) ls += sv[i];
    float wsum = wave_sum_wmma(ls);
    int wid = tid >> 5, lane = tid & 31;
    if (lane == 0) wpart[wid] = wsum;
    __syncthreads();
    if (tid == 0) {
      float tot = 0.0f;
      for (int w = 0; w < 8; ++w) tot += wpart[w];
      atomicAdd(&acc[2], tot);
    }
    return;
  }
  u32 cur = 0;                               // all values >= 0 -> bits monotonic
  for (int bit = 30; bit >= 0; --bit) {
    u32 cand = cur | (1u << bit);
    if (tid == 0) scnt = 0;
    __syncthreads();
    int lc = 0;
    for (int i = tid; i < P_; i += 256) lc += (__float_as_uint(sv[i]) >= cand) ? 1 : 0;
    atomicAdd(&scnt, lc);
    __syncthreads();
    if (scnt >= K) cur = cand;               // uniform decision (shared scnt)
    __syncthreads();
  }
  float t = __uint_as_float(cur);            // K-th largest value (exists by construction)
  if (tid == 0) { scnt = 0; ssum = 0.0f; }
  __syncthreads();
  int lc = 0; float ls = 0.0f;
  for (int i = tid; i < P_; i += 256) {
    float v = sv[i];
    if (v > t) { lc++; ls += v; }
  }
  atomicAdd(&scnt, lc);
  atomicAdd(&ssum, ls);
  __syncthreads();
  if (tid == 0) {
    float total = ssum + (float)(K - (long long)scnt) * t;  // remaining selected all equal t
    atomicAdd(&acc[2], total);
  }
}

__global__ void final_kernel(const int* __restrict__ npos,
                             const float* __restrict__ acc,
                             float* __restrict__ out) {
  if (threadIdx.x == 0 && blockIdx.x == 0) {
    long long npt = 0;
    for (int i = 0; i < B_; ++i) npt += npos[i];
    float conf = (acc[1] + acc[2]) / (float)npt;
    long long d = npt * 4; if (d < 1) d = 1;
    float loc = acc[0] / (float)d;
    out[0] = conf + loc;
  }
}

extern "C" void kernel_launch(void* const* d_in, const int* in_sizes, int n_in,
                              void* d_out, int out_size, void* d_ws, size_t ws_size,
                              hipStream_t stream) {
  const float* ploc   = (const float*)d_in[0];  // (B,P,4)
  const float* scores = (const float*)d_in[1];  // (B,P,C)
  const float* boxes  = (const float*)d_in[2];  // (B,M,4)
  const int*   labels = (const int*)  d_in[3];  // (B,M)
  const float* priors = (const float*)d_in[4];  // (P,4)
  float* out = (float*)d_out;

  char* ws = (char*)d_ws;
  float* ovl   = (float*)ws;  ws += sizeof(float) * BP_;
  int*   obj   = (int*)ws;    ws += sizeof(int)   * BP_;
  int*   tc    = (int*)ws;    ws += sizeof(int)   * BP_;
  float* ceneg = (float*)ws;  ws += sizeof(float) * BP_;
  u64*   bestp = (u64*)ws;    ws += sizeof(u64)   * B_ * M_;
  int*   npos  = (int*)ws;    ws += sizeof(int)   * B_;
  float* acc   = (float*)ws;  ws += sizeof(float) * 8;
  (void)in_sizes; (void)n_in; (void)out_size; (void)ws_size;

  init_ws<<<dim3((B_ * M_ + 255) / 256), 256, 0, stream>>>(bestp, npos, acc);
  overlap_kernel<<<dim3((P_ + 255) / 256, B_), 256, 0, stream>>>(boxes, priors, ovl, obj, bestp);
  override_kernel<<<dim3((B_ * M_ + 255) / 256), 256, 0, stream>>>(ovl, obj, bestp);
  match_loc_kernel<<<dim3((P_ + 255) / 256, B_), 256, 0, stream>>>(
      ploc, boxes, labels, priors, ovl, obj, tc, npos, acc);
  ce_kernel<<<dim3(BP_ / 8), 256, 0, stream>>>(scores, tc, ceneg, acc);
  hardneg_kernel<<<dim3(B_), 256, 0, stream>>>(ceneg, npos, acc);
  final_kernel<<<dim3(1), 64, 0, stream>>>(npos, acc, out);
}